// OrbMessage_65335042506808
// MI455X (gfx1250) — compile-verified
//
#include <hip/hip_runtime.h>
#include <stdint.h>

// OrbMessage for MI455X (gfx1250, wave32).
//   gate = sigmoid(pair @ W); msg = gate * pair_update * mask * envelope(dist)
//   out[neighbours[n,k], :] += msg[n,k,:]
// GEMM on the matrix pipe via V_WMMA_F32_16X16X4_F32 (fp32 in, full precision).
// pair tiles double-buffered in LDS via GLOBAL_LOAD_ASYNC_TO_LDS_B128
// (ASYNCcnt), so HBM latency for tile i+1 overlaps the WMMAs of tile i.
// W held in 64 VGPRs per wave; scatter via global_atomic_add_f32.

typedef __attribute__((ext_vector_type(2))) float v2f;
typedef __attribute__((ext_vector_type(8))) float v8f;

#define CDIM   128   // pair feature dim (K of GEMM)
#define SDIM   128   // output size     (N of GEMM)
#define KNBR   32    // neighbours per node
#define TILE_M 16    // rows per tile
#define PITCH  132   // LDS row pitch in floats: 132%64==4 -> 32 disjoint
                     // bank-pairs for the per-lane ds_load_b64 A reads;
                     // 132*4 bytes keeps rows 16B-aligned for b128 fills.

// Async global->LDS copy of one 16x128 f32 tile (8 KB): 256 threads x 2 b128.
__device__ __forceinline__ void async_tile_load(const float* __restrict__ src,
                                                float* dstLds, int tid) {
#pragma clang loop unroll(full)
  for (int j = 0; j < 2; ++j) {
    const int idx = tid + j * 256;            // float4 index within tile
    const int e   = idx * 4;
    const int row = e >> 7;                   // /128 (b128 never crosses a row)
    const int c0  = e & 127;
    const uint32_t ldsoff = (uint32_t)(uintptr_t)(dstLds + row * PITCH + c0);
    const uint64_t gaddr  = (uint64_t)(uintptr_t)(src + e);
    asm volatile("global_load_async_to_lds_b128 %0, %1, off"
                 :: "v"(ldsoff), "v"(gaddr) : "memory");
  }
}

__launch_bounds__(256)
__global__ void orb_message_kernel(const float* __restrict__ pair,
                                   const float* __restrict__ pair_update,
                                   const int*   __restrict__ neighbours,
                                   const uint8_t* __restrict__ pair_mask,
                                   const float* __restrict__ distance,
                                   const float* __restrict__ Wmat,
                                   float* __restrict__ out,
                                   int NK) {
  __shared__ __align__(16) float sA[2][TILE_M * PITCH];   // double-buffered tile
  __shared__ float sScale[2][TILE_M];
  __shared__ int   sNeigh[2][TILE_M];

  const int tid   = threadIdx.x;
  const int lane  = tid & 31;
  const int wave  = tid >> 5;        // 8 waves -> 8 column tiles of 16
  const int m     = lane & 15;
  const int halfU = lane >> 4;       // 0: K in {k0,k0+1}; 1: K in {k0+2,k0+3}
  const int gcol  = wave * 16 + m;   // global output column (B/C/D: N = lane&15)

  // ---- Preload this wave's 128x16 slab of W as B fragments (64 VGPRs). ----
  // B 4x16 layout for V_WMMA_F32_16X16X4_F32: lanes 0-15 K=k0,k0+1 (v0,v1),
  // lanes 16-31 K=k0+2,k0+3. Loaded once per block, reused for ~25 tiles.
  v2f bfrag[CDIM / 4];
#pragma clang loop unroll(full)
  for (int kk = 0; kk < CDIM / 4; ++kk) {
    const int kb = kk * 4 + 2 * halfU;
    bfrag[kk].x = Wmat[kb * SDIM + gcol];
    bfrag[kk].y = Wmat[(kb + 1) * SDIM + gcol];
  }

  const int ntiles = NK / TILE_M;    // NK = 1.6M, exactly divisible by 16
  const int stride = gridDim.x;
  int tile = blockIdx.x;
  int buf  = 0;

  // ---- Prologue: prefetch first tile + metadata into buffer 0. ----
  if (tile < ntiles) {
    async_tile_load(pair + (size_t)tile * TILE_M * CDIM, sA[0], tid);
    if (tid < TILE_M) {
      const int row = tile * TILE_M + tid;
      const int n   = row >> 5;               // row / KNBR
      const float d = distance[n];
      const float x  = d * (1.0f / 6.0f);     // r_max = 6, p = 3
      const float x2 = x * x;
      const float x3 = x2 * x;
      float env = 1.0f - 10.0f * x3 + 15.0f * x3 * x - 6.0f * x3 * x2;
      env = (d < 6.0f) ? env : 0.0f;
      sScale[0][tid] = pair_mask[row] ? env : 0.0f;
      sNeigh[0][tid] = neighbours[row];
    }
  }

  for (; tile < ntiles; tile += stride, buf ^= 1) {
    const int rowbase = tile * TILE_M;

    // Wave's own async fills must land before signaling; barrier publishes
    // the tile (and retires every wave's LDS reads of the other buffer).
    asm volatile("s_wait_asynccnt 0" ::: "memory");
    __syncthreads();

    // ---- Prefetch tile i+1 into the other buffer (overlaps the WMMAs). ----
    const int nt = tile + stride;
    if (nt < ntiles) {
      async_tile_load(pair + (size_t)nt * TILE_M * CDIM, sA[buf ^ 1], tid);
      if (tid < TILE_M) {
        const int row = nt * TILE_M + tid;
        const int n   = row >> 5;
        const float d = distance[n];
        const float x  = d * (1.0f / 6.0f);
        const float x2 = x * x;
        const float x3 = x2 * x;
        float env = 1.0f - 10.0f * x3 + 15.0f * x3 * x - 6.0f * x3 * x2;
        env = (d < 6.0f) ? env : 0.0f;
        sScale[buf ^ 1][tid] = pair_mask[row] ? env : 0.0f;
        sNeigh[buf ^ 1][tid] = neighbours[row];
      }
    }

    // ---- 16x16 tile of pair@W via 32 chained V_WMMA_F32_16X16X4_F32. ----
    // A 16x4 layout: lane m (both halves) holds row M=m; v0=A[m][k0+2*half],
    // v1=A[m][k0+2*half+1] -> one aligned 8-byte LDS read per step.
    v8f acc = {};
#pragma clang loop unroll(full)
    for (int kk = 0; kk < CDIM / 4; ++kk) {
      const int koff = kk * 4 + 2 * halfU;    // even -> 8-byte aligned
      const v2f a = *(const v2f*)&sA[buf][m * PITCH + koff];
      acc = __builtin_amdgcn_wmma_f32_16x16x4_f32(
          false, a, false, bfrag[kk], (short)0, acc, false, false);
    }

    // ---- Epilogue: sigmoid * pair_update * scale, scatter-add. ----
    // C/D layout: VGPR r -> M = 8*half + r, N = lane&15.
    const int m0 = halfU * 8;
#pragma clang loop unroll(full)
    for (int r = 0; r < 8; ++r) {
      const int lr = m0 + r;
      const float s = sScale[buf][lr];
      if (s != 0.0f) {
        const float gate = 1.0f / (1.0f + __expf(-acc[r]));
        const float pu = pair_update[(size_t)(rowbase + lr) * SDIM + gcol];
        atomicAdd(&out[(size_t)sNeigh[buf][lr] * SDIM + gcol], gate * pu * s);
      }
    }
  }
}

extern "C" void kernel_launch(void* const* d_in, const int* in_sizes, int n_in,
                              void* d_out, int out_size, void* d_ws, size_t ws_size,
                              hipStream_t stream) {
  const float*   pair        = (const float*)d_in[0];
  const float*   pair_update = (const float*)d_in[1];
  const int*     neighbours  = (const int*)d_in[2];
  const uint8_t* pair_mask   = (const uint8_t*)d_in[3];
  const float*   distance    = (const float*)d_in[4];
  const float*   Wmat        = (const float*)d_in[5];
  float*         out         = (float*)d_out;

  const int NK = in_sizes[2];          // N*K rows (1,600,000)

  // Output is scatter-add accumulated: zero it first (capture-safe memset).
  hipMemsetAsync(d_out, 0, (size_t)out_size * sizeof(float), stream);

  int ntiles = NK / TILE_M;
  int grid = ntiles < 4096 ? ntiles : 4096;   // each block loops over tiles
  orb_message_kernel<<<dim3(grid), dim3(256), 0, stream>>>(
      pair, pair_update, neighbours, pair_mask, distance, Wmat, out, NK);
}